// ResNetPostMP_1881195675758
// MI455X (gfx1250) — compile-verified
//
#include <hip/hip_runtime.h>
#include <cstdint>
#include <cstddef>

#define N_NODES 100000
#define N_EDGES 1600000
#define DIN_F   100
#define HDIM    128
#define PHDIM   256
#define DOUT_F  47

typedef __attribute__((ext_vector_type(2))) float v2f;
typedef __attribute__((ext_vector_type(8))) float v8f;

// ---------------------------------------------------------------------------
// fp32 WMMA: D = A(16x4) x B(4x16) + C(16x16), exact fp32 semantics.
// ---------------------------------------------------------------------------
__device__ __forceinline__ v8f wmma4(v2f a, v2f b, v8f c) {
  return __builtin_amdgcn_wmma_f32_16x16x4_f32(false, a, false, b, (short)0, c,
                                               false, false);
}

// ---------------------------------------------------------------------------
// Stage a weight matrix [nrows x K] (row-major, out_features-major) into LDS
// pre-swizzled into WMMA B-fragment layout:
//   lds[(ks*NTILES + t)*64 + lane*2 + j] = W[t*16 + (lane&15)][4*ks + 2*(lane>>4) + j]
// Rows >= nrows are zero-filled (used to pad the 47-col final layer).
// All threads of the block participate (call before a barrier).
// ---------------------------------------------------------------------------
template <int NTILES, int K>
__device__ __forceinline__ void stage_swizzled(const float* __restrict__ W, int nrows,
                                               float* __restrict__ lds, int tid, int nthr) {
  constexpr int TOT = NTILES * 16 * K;
  for (int i = 4 * tid; i < TOT; i += 4 * nthr) {
    const int n  = i / K;
    const int kk = i - n * K;           // K % 4 == 0 -> kk % 4 == 0
    float4 v;
    if (n < nrows) v = *(const float4*)(W + (size_t)n * K + kk);
    else           v = make_float4(0.f, 0.f, 0.f, 0.f);
    const int ks  = kk >> 2;
    const int t   = n >> 4;
    const int l   = n & 15;
    float* d = lds + (size_t)(ks * NTILES + t) * 64;
    d[2 * l]          = v.x;            // lane l      (half 0), j=0
    d[2 * l + 1]      = v.y;            // lane l      (half 0), j=1
    d[32 + 2 * l]     = v.z;            // lane 16+l   (half 1), j=0
    d[32 + 2 * l + 1] = v.w;            // lane 16+l   (half 1), j=1
  }
}

// ---------------------------------------------------------------------------
// Elementwise / scatter utility kernels
// ---------------------------------------------------------------------------
__global__ void fill_zero_k(float* __restrict__ p, size_t n) {
  size_t i = (size_t)blockIdx.x * blockDim.x + threadIdx.x;
  if (i < n) p[i] = 0.0f;
}

__global__ void count_k(const int* __restrict__ dst, float* __restrict__ cnt, int E) {
  int e = blockIdx.x * blockDim.x + threadIdx.x;
  if (e < E) atomicAdd(&cnt[dst[e]], 1.0f);
}

__global__ void invcnt_k(float* __restrict__ cnt, int n) {
  int i = blockIdx.x * blockDim.x + threadIdx.x;
  if (i >= n) return;
  float c = cnt[i];
  cnt[i] = (c > 0.0f) ? (1.0f / fmaxf(c, 1.0f)) : 0.0f;
}

// per-edge gather + atomic scatter-add, float4 feature chunks (d % 4 == 0)
__global__ void scatter_k(const float* __restrict__ x, const int* __restrict__ src,
                          const int* __restrict__ dst, float* __restrict__ s,
                          int E, int d) {
  const int chunks = d >> 2;
  int idx = blockIdx.x * blockDim.x + threadIdx.x;
  if (idx >= E * chunks) return;
  int e = idx / chunks;
  int c = (idx - e * chunks) << 2;
  const float4 v = *(const float4*)(x + (size_t)src[e] * d + c);
  float* p = s + (size_t)dst[e] * d + c;
  atomicAdd(p + 0, v.x);
  atomicAdd(p + 1, v.y);
  atomicAdd(p + 2, v.z);
  atomicAdd(p + 3, v.w);
}

// s[i,:] *= inv_cnt[i]  (float4 vectorized)
__global__ void scale_rows_k(float* __restrict__ s, const float* __restrict__ inv,
                             int n, int d) {
  const int chunks = d >> 2;
  int idx = blockIdx.x * blockDim.x + threadIdx.x;
  if (idx >= n * chunks) return;
  int i = idx / chunks;
  int c = (idx - i * chunks) << 2;
  float4* p = (float4*)(s + (size_t)i * d + c);
  float4 v = *p;
  float f = inv[i];
  v.x *= f; v.y *= f; v.z *= f; v.w *= f;
  *p = v;
}

// ---------------------------------------------------------------------------
// SAGE fused kernel: out = relu(L2normalize_row(X@Wl^T + bl + Z@Wr^T + br)).
// One wave owns a 16-row x 128-col strip (8 v8f accumulators).
// Weights staged in LDS (fragment layout), two passes: Wl then Wr.
// ---------------------------------------------------------------------------
template <int K>
__global__ void __launch_bounds__(256)
sage_wmma_k(const float* __restrict__ X, const float* __restrict__ Z,
            const float* __restrict__ Wl, const float* __restrict__ Bl,
            const float* __restrict__ Wr, const float* __restrict__ Br,
            float* __restrict__ out, int M) {
  constexpr int NT = 8;                           // 8 col tiles = 128 cols
  __shared__ float lw[(K / 4) * NT * 64];         // K*128 floats (<= 64 KB)

  const int lane = threadIdx.x & 31;
  const int wave = threadIdx.x >> 5;
  const int half = lane >> 4;
  const int l16  = lane & 15;
  const int rowTiles = M >> 4;
  const int tileIdx = blockIdx.x * (blockDim.x >> 5) + wave;
  const bool active = tileIdx < rowTiles;
  const int row0 = (active ? tileIdx : 0) << 4;   // clamped; inactive waves skip stores

  v8f acc[NT] = {};
  const float* bbase = lw + lane * 2;             // fragment base (imm offsets below)

  // ---- pass 1: X @ Wl^T ----
  stage_swizzled<NT, K>(Wl, NT * 16, lw, threadIdx.x, blockDim.x);
  __syncthreads();
  {
    const float* ax = X + (size_t)(row0 + l16) * K + 2 * half;
#pragma unroll
    for (int ks = 0; ks < K / 4; ++ks) {
      const v2f a = *(const v2f*)(ax + 4 * ks);
#pragma unroll
      for (int t = 0; t < NT; ++t) {
        const v2f b = *(const v2f*)(bbase + (ks * NT + t) * 64);
        acc[t] = wmma4(a, b, acc[t]);
      }
    }
  }

  // ---- pass 2: Z @ Wr^T (restage LDS) ----
  __syncthreads();
  stage_swizzled<NT, K>(Wr, NT * 16, lw, threadIdx.x, blockDim.x);
  __syncthreads();
  {
    const float* az = Z + (size_t)(row0 + l16) * K + 2 * half;
#pragma unroll
    for (int ks = 0; ks < K / 4; ++ks) {
      const v2f a = *(const v2f*)(az + 4 * ks);
#pragma unroll
      for (int t = 0; t < NT; ++t) {
        const v2f b = *(const v2f*)(bbase + (ks * NT + t) * 64);
        acc[t] = wmma4(a, b, acc[t]);
      }
    }
  }

  // ---- epilogue: bias, row L2 norm, relu ----
  float bsum[NT];
#pragma unroll
  for (int t = 0; t < NT; ++t) bsum[t] = Bl[t * 16 + l16] + Br[t * 16 + l16];

#pragma unroll
  for (int r = 0; r < 8; ++r) {
    float vals[NT];
    float ss = 0.0f;
#pragma unroll
    for (int t = 0; t < NT; ++t) {
      vals[t] = acc[t][r] + bsum[t];
      ss += vals[t] * vals[t];
    }
    ss += __shfl_xor(ss, 1, 32);
    ss += __shfl_xor(ss, 2, 32);
    ss += __shfl_xor(ss, 4, 32);
    ss += __shfl_xor(ss, 8, 32);
    const float invn = 1.0f / fmaxf(sqrtf(ss), 1e-12f);
    if (active) {
      const size_t ro = (size_t)(row0 + r + 8 * half) * HDIM;
#pragma unroll
      for (int t = 0; t < NT; ++t)
        out[ro + t * 16 + l16] = fmaxf(vals[t] * invn, 0.0f);
    }
  }
}

// ---------------------------------------------------------------------------
// Generic linear into PHDIM=256 cols: out = ACT(X@W^T + bias) [+ res].
// Wave owns 16x64 tile; block stages its 64 output rows of W in LDS.
// ---------------------------------------------------------------------------
template <int K, int ACT, bool RES>
__global__ void __launch_bounds__(256)
linear_wmma_k(const float* __restrict__ X, const float* __restrict__ W,
              const float* __restrict__ bias, const float* __restrict__ res,
              float* __restrict__ out, int M) {
  constexpr int NT = 4;                           // 64 cols per block
  __shared__ float lw[(K / 4) * NT * 64];         // K*64 floats (<= 64 KB)

  const int lane = threadIdx.x & 31;
  const int wave = threadIdx.x >> 5;
  const int half = lane >> 4;
  const int l16  = lane & 15;
  const int rowTiles = M >> 4;
  const int tileIdx = blockIdx.x * (blockDim.x >> 5) + wave;
  const bool active = tileIdx < rowTiles;
  const int row0 = (active ? tileIdx : 0) << 4;
  const int n0 = blockIdx.y * 64;

  stage_swizzled<NT, K>(W + (size_t)n0 * K, NT * 16, lw, threadIdx.x, blockDim.x);
  __syncthreads();

  v8f acc[NT] = {};
  const float* bbase = lw + lane * 2;
  const float* ax = X + (size_t)(row0 + l16) * K + 2 * half;
#pragma unroll
  for (int ks = 0; ks < K / 4; ++ks) {
    const v2f a = *(const v2f*)(ax + 4 * ks);
#pragma unroll
    for (int t = 0; t < NT; ++t) {
      const v2f b = *(const v2f*)(bbase + (ks * NT + t) * 64);
      acc[t] = wmma4(a, b, acc[t]);
    }
  }

  if (!active) return;                            // no more barriers below
#pragma unroll
  for (int t = 0; t < NT; ++t) {
    const float bv = bias[n0 + t * 16 + l16];
#pragma unroll
    for (int r = 0; r < 8; ++r) {
      float v = acc[t][r] + bv;
      if (ACT == 1) v = fmaxf(v, 0.0f);
      const size_t o = (size_t)(row0 + r + 8 * half) * PHDIM + n0 + t * 16 + l16;
      if (RES) v += res[o];
      out[o] = v;
    }
  }
}

// ---------------------------------------------------------------------------
// BatchNorm: column stats (atomic partial sums), finalize, apply.
// ---------------------------------------------------------------------------
__global__ void bn_stats_k(const float* __restrict__ t, float* __restrict__ sums, int M) {
  const int col = threadIdx.x;  // 256 threads == PHDIM columns
  const int row0 = blockIdx.x * 256;
  const int rend = (M - row0 < 256) ? (M - row0) : 256;
  float s = 0.0f, q = 0.0f;
  for (int r = 0; r < rend; ++r) {
    float v = t[(size_t)(row0 + r) * PHDIM + col];
    s += v;
    q += v * v;
  }
  atomicAdd(&sums[col], s);
  atomicAdd(&sums[PHDIM + col], q);
}

__global__ void bn_finalize_k(const float* __restrict__ sums, const float* __restrict__ g,
                              const float* __restrict__ beta, float* __restrict__ ss, int M) {
  const int c = threadIdx.x;
  const float mean = sums[c] / (float)M;
  const float var  = sums[PHDIM + c] / (float)M - mean * mean;
  const float sc   = g[c] * rsqrtf(var + 1e-5f);
  ss[c]         = sc;
  ss[PHDIM + c] = beta[c] - mean * sc;
}

__global__ void bn_apply_k(float* __restrict__ t, const float* __restrict__ ss, size_t n) {
  size_t i = (size_t)blockIdx.x * blockDim.x + threadIdx.x;
  if (i >= n) return;
  const int c = (int)(i & (PHDIM - 1));
  t[i] = t[i] * ss[c] + ss[PHDIM + c];
}

__global__ void bn_apply_res_k(const float* __restrict__ t, const float* __restrict__ ss,
                               float* __restrict__ h, size_t n) {
  size_t i = (size_t)blockIdx.x * blockDim.x + threadIdx.x;
  if (i >= n) return;
  const int c = (int)(i & (PHDIM - 1));
  h[i] = h[i] + t[i] * ss[c] + ss[PHDIM + c];
}

// ---------------------------------------------------------------------------
// Final: logits = X@W^T + b (47 cols; staged zero-padded to 48), log_softmax.
// ---------------------------------------------------------------------------
__global__ void __launch_bounds__(256)
final_lsm_wmma_k(const float* __restrict__ X, const float* __restrict__ W,
                 const float* __restrict__ bias, float* __restrict__ out, int M) {
  constexpr int K  = PHDIM;                        // 256
  constexpr int NT = 3;                            // 48 cols (47 valid + 1 pad)
  __shared__ float lw[(K / 4) * NT * 64];          // 48 KB

  const int lane = threadIdx.x & 31;
  const int wave = threadIdx.x >> 5;
  const int half = lane >> 4;
  const int l16  = lane & 15;
  const int rowTiles = M >> 4;
  const int tileIdx = blockIdx.x * (blockDim.x >> 5) + wave;
  const bool active = tileIdx < rowTiles;
  const int row0 = (active ? tileIdx : 0) << 4;

  stage_swizzled<NT, K>(W, DOUT_F, lw, threadIdx.x, blockDim.x);  // row 47 -> zeros
  __syncthreads();

  v8f acc[NT] = {};
  const float* bbase = lw + lane * 2;
  const float* ax = X + (size_t)(row0 + l16) * K + 2 * half;
#pragma unroll
  for (int ks = 0; ks < K / 4; ++ks) {
    const v2f a = *(const v2f*)(ax + 4 * ks);
#pragma unroll
    for (int t = 0; t < NT; ++t) {
      const v2f b = *(const v2f*)(bbase + (ks * NT + t) * 64);
      acc[t] = wmma4(a, b, acc[t]);
    }
  }
  if (!active) return;

  int  cn[NT];
  bool ok[NT];
  float bv[NT];
#pragma unroll
  for (int t = 0; t < NT; ++t) {
    cn[t] = t * 16 + l16;
    ok[t] = cn[t] < DOUT_F;
    bv[t] = ok[t] ? bias[cn[t]] : 0.0f;
  }

#pragma unroll
  for (int r = 0; r < 8; ++r) {
    float v[NT];
#pragma unroll
    for (int t = 0; t < NT; ++t) v[t] = ok[t] ? (acc[t][r] + bv[t]) : -INFINITY;

    float mx = fmaxf(fmaxf(v[0], v[1]), v[2]);
    mx = fmaxf(mx, __shfl_xor(mx, 1, 32));
    mx = fmaxf(mx, __shfl_xor(mx, 2, 32));
    mx = fmaxf(mx, __shfl_xor(mx, 4, 32));
    mx = fmaxf(mx, __shfl_xor(mx, 8, 32));

    float se = 0.0f;
#pragma unroll
    for (int t = 0; t < NT; ++t) if (ok[t]) se += expf(v[t] - mx);
    se += __shfl_xor(se, 1, 32);
    se += __shfl_xor(se, 2, 32);
    se += __shfl_xor(se, 4, 32);
    se += __shfl_xor(se, 8, 32);
    const float lse = logf(se);

    const size_t ro = (size_t)(row0 + r + 8 * half) * DOUT_F;
#pragma unroll
    for (int t = 0; t < NT; ++t)
      if (ok[t]) out[ro + cn[t]] = v[t] - mx - lse;
  }
}

// ---------------------------------------------------------------------------
// Host-side orchestration
// ---------------------------------------------------------------------------
extern "C" void kernel_launch(void* const* d_in, const int* in_sizes, int n_in,
                              void* d_out, int out_size, void* d_ws, size_t ws_size,
                              hipStream_t stream) {
  (void)in_sizes; (void)n_in; (void)out_size; (void)ws_size;

  const float* x   = (const float*)d_in[0];
  const int*   ei  = (const int*)d_in[1];
  const int*   src = ei;
  const int*   dst = ei + N_EDGES;
  auto F = [&](int i) { return (const float*)d_in[i]; };

  // workspace layout
  float* ws    = (float*)d_ws;
  float* cnt   = ws;                                    // N
  float* sbuf  = cnt  + N_NODES;                        // N*128
  float* bufA  = sbuf + (size_t)N_NODES * HDIM;         // N*256
  float* bufB  = bufA + (size_t)N_NODES * PHDIM;        // N*256
  float* bnsum = bufB + (size_t)N_NODES * PHDIM;        // 512
  float* bnss  = bnsum + 2 * PHDIM;                     // 512

  const int THR = 256;
  const int rowTiles = N_NODES / 16;                    // 6250
  const dim3 gRows((rowTiles + 7) / 8);                 // 8 waves per block
  const dim3 gLin((rowTiles + 7) / 8, PHDIM / 64);      // 16x64 tiles, Nout=256

  // --- degree counts -> inverse counts (in place) ---
  fill_zero_k<<<(N_NODES + THR - 1) / THR, THR, 0, stream>>>(cnt, N_NODES);
  count_k<<<(N_EDGES + THR - 1) / THR, THR, 0, stream>>>(dst, cnt, N_EDGES);
  invcnt_k<<<(N_NODES + THR - 1) / THR, THR, 0, stream>>>(cnt, N_NODES);

  // --- one SAGE conv layer ---
  auto conv_pre = [&](const float* hin, int din) {
    const size_t sn = (size_t)N_NODES * din;
    fill_zero_k<<<(unsigned)((sn + THR - 1) / THR), THR, 0, stream>>>(sbuf, sn);
    const int chunks = din >> 2;
    const int totE = N_EDGES * chunks;
    scatter_k<<<(totE + THR - 1) / THR, THR, 0, stream>>>(hin, src, dst, sbuf,
                                                          N_EDGES, din);
    const int totN = N_NODES * chunks;
    scale_rows_k<<<(totN + THR - 1) / THR, THR, 0, stream>>>(sbuf, cnt, N_NODES, din);
  };

  // conv0: x -> bufA (K = 100)
  conv_pre(x, DIN_F);
  sage_wmma_k<DIN_F><<<gRows, THR, 0, stream>>>(x, sbuf, F(2), F(3), F(4), F(5),
                                                bufA, N_NODES);
  // conv1: bufA -> bufB (K = 128)
  conv_pre(bufA, HDIM);
  sage_wmma_k<HDIM><<<gRows, THR, 0, stream>>>(bufA, sbuf, F(6), F(7), F(8), F(9),
                                               bufB, N_NODES);
  // conv2: bufB -> bufA (K = 128)
  conv_pre(bufB, HDIM);
  sage_wmma_k<HDIM><<<gRows, THR, 0, stream>>>(bufB, sbuf, F(10), F(11), F(12), F(13),
                                               bufA, N_NODES);

  // --- post_mp: p0 (128 -> 256), no activation ---
  linear_wmma_k<HDIM, 0, false><<<gLin, THR, 0, stream>>>(bufA, F(14), F(15), nullptr,
                                                          bufB, N_NODES);

  const size_t nPH = (size_t)N_NODES * PHDIM;
  const unsigned gEl = (unsigned)((nPH + THR - 1) / THR);
  const unsigned gBN = (N_NODES + 255) / 256;

  // --- residual block: h(bufB) -> t(bufA) -> BN -> GEMM + residual (in-place bufB)
  auto resblock = [&](const float* w1, const float* b1, const float* g,
                      const float* beta, const float* w2, const float* b2) {
    linear_wmma_k<PHDIM, 1, false><<<gLin, THR, 0, stream>>>(bufB, w1, b1, nullptr,
                                                             bufA, N_NODES);
    fill_zero_k<<<2, THR, 0, stream>>>(bnsum, 2 * PHDIM);
    bn_stats_k<<<gBN, PHDIM, 0, stream>>>(bufA, bnsum, N_NODES);
    bn_finalize_k<<<1, PHDIM, 0, stream>>>(bnsum, g, beta, bnss, N_NODES);
    bn_apply_k<<<gEl, THR, 0, stream>>>(bufA, bnss, nPH);
    linear_wmma_k<PHDIM, 0, true><<<gLin, THR, 0, stream>>>(bufA, w2, b2, bufB,
                                                            bufB, N_NODES);
  };
  resblock(F(16), F(17), F(18), F(19), F(20), F(21));    // b1
  resblock(F(22), F(23), F(24), F(25), F(26), F(27));    // b2

  // --- block 3: Linear -> ReLU -> BN, residual add (bufB += BN(t)) ---
  linear_wmma_k<PHDIM, 1, false><<<gLin, THR, 0, stream>>>(bufB, F(28), F(29), nullptr,
                                                           bufA, N_NODES);
  fill_zero_k<<<2, THR, 0, stream>>>(bnsum, 2 * PHDIM);
  bn_stats_k<<<gBN, PHDIM, 0, stream>>>(bufA, bnsum, N_NODES);
  bn_finalize_k<<<1, PHDIM, 0, stream>>>(bnsum, F(30), F(31), bnss, N_NODES);
  bn_apply_res_k<<<gEl, THR, 0, stream>>>(bufA, bnss, bufB, nPH);

  // --- final 256 -> 47 + log_softmax ---
  final_lsm_wmma_k<<<gRows, THR, 0, stream>>>(bufB, F(32), F(33), (float*)d_out,
                                              N_NODES);
}